// FFTConv_33432025432208
// MI455X (gfx1250) — compile-verified
//
#include <hip/hip_runtime.h>
#include <hip/hip_bf16.h>

// FFT conv: y = irfft2( einsum('oihw,bihw->bohw', K, rfft2(x)) ) + bias
// B=8, Cin=Cout=32, H=256, W=256, WF=129.
// Memory-bound (~500-700MB @ 23.3TB/s). f32 WMMA (16x16x4) used for the
// per-frequency complex channel-mix GEMM.

#define Hn   256
#define Wn   256
#define WF   129
#define HWF  (Hn * WF)          // 33024 frequencies
#define Bn   8
#define CIN  32
#define COUT 32

typedef __attribute__((ext_vector_type(2))) float v2f;
typedef __attribute__((ext_vector_type(8))) float v8f;

__device__ __forceinline__ float2 cmulf(float2 a, float2 b) {
    return make_float2(a.x * b.x - a.y * b.y, a.x * b.y + a.y * b.x);
}

// ---------------------------------------------------------------------------
// 256-point radix-4 Stockham FFT in LDS (ping-pong). 64 threads per FFT,
// thread j does one radix-4 butterfly per stage. dir = -1 fwd, +1 inverse.
// Result ends in buffer `a`. Ends with a barrier.
// ---------------------------------------------------------------------------
__device__ __forceinline__ void fft256(float2* a, float2* b, int j, float dir) {
    float2* src = a;
    float2* dst = b;
#pragma unroll
    for (int stage = 0; stage < 4; ++stage) {
        const int Ns = 1 << (2 * stage);
        const int r  = j & (Ns - 1);
        float2 u0 = src[j];
        float2 u1 = src[j + 64];
        float2 u2 = src[j + 128];
        float2 u3 = src[j + 192];
        float ang = dir * 6.283185307179586f * (float)r / (float)(4 * Ns);
        float s, c;
        __sincosf(ang, &s, &c);
        float2 w1 = make_float2(c, s);
        float2 w2 = cmulf(w1, w1);
        float2 w3 = cmulf(w2, w1);
        u1 = cmulf(u1, w1);
        u2 = cmulf(u2, w2);
        u3 = cmulf(u3, w3);
        float2 t0  = make_float2(u0.x + u2.x, u0.y + u2.y);
        float2 t1  = make_float2(u0.x - u2.x, u0.y - u2.y);
        float2 t2  = make_float2(u1.x + u3.x, u1.y + u3.y);
        float2 d13 = make_float2(u1.x - u3.x, u1.y - u3.y);
        float2 t3  = make_float2(-dir * d13.y, dir * d13.x);  // dir*i*d13
        int base = ((j >> (2 * stage)) << (2 * stage + 2)) + r;
        dst[base]          = make_float2(t0.x + t2.x, t0.y + t2.y);
        dst[base + Ns]     = make_float2(t1.x + t3.x, t1.y + t3.y);
        dst[base + 2 * Ns] = make_float2(t0.x - t2.x, t0.y - t2.y);
        dst[base + 3 * Ns] = make_float2(t1.x - t3.x, t1.y - t3.y);
        __syncthreads();
        float2* tmp = src; src = dst; dst = tmp;
    }
}

// ---------------------------------------------------------------------------
// Stage 1: rfft along W for every (b, cin, h) row. 4 rows per 256-thr block.
// Output planes Xre/Xim, layout [bc][h*WF + w], bc = b*CIN + i.
// ---------------------------------------------------------------------------
__global__ void rfft_rows(const float* __restrict__ x,
                          float* __restrict__ Xre, float* __restrict__ Xim) {
    __shared__ float2 bufA[4][256];
    __shared__ float2 bufB[4][256];
    const int tid = threadIdx.x;
    const int q = tid >> 6;
    const int j = tid & 63;
    const long row = (long)blockIdx.x * 4 + q;          // 0 .. B*CIN*H-1
    const float* src = x + row * Wn;
#pragma unroll
    for (int k = 0; k < 4; ++k) {
        int e = j + 64 * k;
        bufA[q][e] = make_float2(src[e], 0.0f);
    }
    __syncthreads();
    fft256(&bufA[q][0], &bufB[q][0], j, -1.0f);
    const long bc = row >> 8;
    const int h = (int)(row & 255);
    float* outRe = Xre + bc * HWF + (long)h * WF;
    float* outIm = Xim + bc * HWF + (long)h * WF;
    float2 v0 = bufA[q][j];
    float2 v1 = bufA[q][j + 64];
    outRe[j] = v0.x;       outIm[j] = v0.y;
    outRe[j + 64] = v1.x;  outIm[j + 64] = v1.y;
    if (j == 0) {
        float2 v2 = bufA[q][128];
        outRe[128] = v2.x; outIm[128] = v2.y;
    }
}

// ---------------------------------------------------------------------------
// Stages 2 & 4: in-place complex FFT along H for 4 spectral columns per block.
// dir = -1 fwd (scale 1), +1 inv (scale 1/256).
// ---------------------------------------------------------------------------
__global__ void fft_cols(float* __restrict__ Xre, float* __restrict__ Xim,
                         float dir, float scale) {
    __shared__ float2 bufA[4][256];
    __shared__ float2 bufB[4][256];
    const int tid = threadIdx.x;
    const int blkW = blockIdx.x % 33;
    const long bc  = blockIdx.x / 33;
    const int w0 = blkW * 4;
    const long base = bc * HWF;
#pragma unroll
    for (int c = 0; c < 4; ++c) {
        int idx = c * 256 + tid;
        int h = idx >> 2, wl = idx & 3;
        int w = w0 + wl;
        int wc = (w > 128) ? 128 : w;                   // clamp (guarded store)
        bufA[wl][h] = make_float2(Xre[base + (long)h * WF + wc],
                                  Xim[base + (long)h * WF + wc]);
    }
    __syncthreads();
    const int q = tid >> 6;
    const int j = tid & 63;
    fft256(&bufA[q][0], &bufB[q][0], j, dir);
#pragma unroll
    for (int c = 0; c < 4; ++c) {
        int idx = c * 256 + tid;
        int h = idx >> 2, wl = idx & 3;
        int w = w0 + wl;
        if (w < WF) {
            float2 v = bufA[wl][h];
            Xre[base + (long)h * WF + w] = v.x * scale;
            Xim[base + (long)h * WF + w] = v.y * scale;
        }
    }
}

// ---------------------------------------------------------------------------
// Stage 3: spectral channel mix with f32 WMMA.  Per frequency f:
//   Y[o,b] = sum_i K[o,i] * X[i,b]   (complex)
// Block = 256 threads = 8 waves, covers 4 frequencies; wave w handles
// frequency fl = w>>1 and output M-tile mt = w&1 (Cout rows mt*16..mt*16+15).
// K tile + X tile staged to LDS (coalesced float4 global reads); Y written
// in place over X (disjoint f-windows across blocks; read-before-write here).
// ---------------------------------------------------------------------------
__device__ __forceinline__ v8f wmma4(v2f a, v2f b, v8f c) {
    return __builtin_amdgcn_wmma_f32_16x16x4_f32(
        false, a, false, b, (short)0, c, false, false);
}

__global__ void spectral_mix(const float* __restrict__ Kre,
                             const float* __restrict__ Kim,
                             float* __restrict__ Xre,
                             float* __restrict__ Xim) {
    // [i][fl][o] with o padded 32->35 : conflict-free operand gathers
    __shared__ float sKre[CIN][4][35];
    __shared__ float sKim[CIN][4][35];
    // [i][fl][n] with n padded 16->19 ; n=8..15 are zeros (B=8 < 16)
    __shared__ float sXre[CIN][4][19];
    __shared__ float sXim[CIN][4][19];

    const int tid = threadIdx.x;
    const int f0 = blockIdx.x * 4;

    // ---- stage K tile: 1024 (o,i) pairs x 4 freqs, float4 per pair ----
#pragma unroll
    for (int p = tid; p < COUT * CIN; p += 256) {
        int i = p & 31, o = p >> 5;
        const float* kr = Kre + (long)(o * CIN + i) * HWF + f0;
        const float* ki = Kim + (long)(o * CIN + i) * HWF + f0;
        __builtin_prefetch(kr + HWF, 0, 1);             // global_prefetch_b8
        float4 r = *(const float4*)kr;
        float4 m = *(const float4*)ki;
        sKre[i][0][o] = r.x; sKre[i][1][o] = r.y;
        sKre[i][2][o] = r.z; sKre[i][3][o] = r.w;
        sKim[i][0][o] = m.x; sKim[i][1][o] = m.y;
        sKim[i][2][o] = m.z; sKim[i][3][o] = m.w;
    }
    // ---- stage X tile: 32 i x 8 b x 4 freqs (+ zero columns 8..15) ----
    {
        int i = tid >> 3, bb = tid & 7;
        const float* xr = Xre + (long)(bb * CIN + i) * HWF + f0;
        const float* xi = Xim + (long)(bb * CIN + i) * HWF + f0;
        float4 r = *(const float4*)xr;
        float4 m = *(const float4*)xi;
        sXre[i][0][bb] = r.x; sXre[i][1][bb] = r.y;
        sXre[i][2][bb] = r.z; sXre[i][3][bb] = r.w;
        sXim[i][0][bb] = m.x; sXim[i][1][bb] = m.y;
        sXim[i][2][bb] = m.z; sXim[i][3][bb] = m.w;
#pragma unroll
        for (int fl = 0; fl < 4; ++fl) {
            sXre[i][fl][bb + 8] = 0.0f;
            sXim[i][fl][bb + 8] = 0.0f;
        }
    }
    __syncthreads();

    const int wid = tid >> 5, lane = tid & 31;
    const int fl = wid >> 1, mt = wid & 1;
    const int f = f0 + fl;
    const int lh = lane >> 4;       // lane half selects k-pair within slice
    const int m  = lane & 15;       // A: row M ; B: col N
    const int oa = mt * 16 + m;

    v8f accRe = {0.f, 0.f, 0.f, 0.f, 0.f, 0.f, 0.f, 0.f};
    v8f accIm = {0.f, 0.f, 0.f, 0.f, 0.f, 0.f, 0.f, 0.f};

#pragma unroll
    for (int kk = 0; kk < 8; ++kk) {                    // Cin = 8 * k4
        const int i0 = kk * 4 + 2 * lh;                 // same slot->k map for A and B
        v2f aRe, aIm, bRe, bIm;
        aRe[0] = sKre[i0][fl][oa];     aRe[1] = sKre[i0 + 1][fl][oa];
        aIm[0] = sKim[i0][fl][oa];     aIm[1] = sKim[i0 + 1][fl][oa];
        bRe[0] = sXre[i0][fl][m];      bRe[1] = sXre[i0 + 1][fl][m];
        bIm[0] = sXim[i0][fl][m];      bIm[1] = sXim[i0 + 1][fl][m];
        v2f bImN = -bIm;
        accRe = wmma4(aRe, bRe, accRe);                 // + Kr*Xr
        accRe = wmma4(aIm, bImN, accRe);                // - Ki*Xi
        accIm = wmma4(aRe, bIm, accIm);                 // + Kr*Xi
        accIm = wmma4(aIm, bRe, accIm);                 // + Ki*Xr
    }

    // ---- store D (16x16 f32 layout: lane n + 16*(m>>3), vgpr m&7) ----
    const int bcol = m;                                 // N column = batch b
    if (bcol < Bn) {
        const int ob = mt * 16 + 8 * lh;
#pragma unroll
        for (int v = 0; v < 8; ++v) {
            long gi = (long)(bcol * COUT + ob + v) * HWF + f;
            Xre[gi] = accRe[v];
            Xim[gi] = accIm[v];
        }
    }
}

// ---------------------------------------------------------------------------
// Stage 5: irfft along W (Hermitian extension 129 -> 256), x 1/256, + bias.
// ---------------------------------------------------------------------------
__global__ void irfft_rows_bias(const float* __restrict__ Yre,
                                const float* __restrict__ Yim,
                                const float* __restrict__ bias,
                                float* __restrict__ out) {
    __shared__ float2 bufA[4][256];
    __shared__ float2 bufB[4][256];
    const int tid = threadIdx.x;
    const int q = tid >> 6;
    const int j = tid & 63;
    const long row = (long)blockIdx.x * 4 + q;          // 0 .. B*COUT*H-1
    const long bc = row >> 8;
    const int h = (int)(row & 255);
    const float* pre = Yre + bc * HWF + (long)h * WF;
    const float* pim = Yim + bc * HWF + (long)h * WF;
#pragma unroll
    for (int k = 0; k < 4; ++k) {
        int e = j + 64 * k;
        if (e <= 128) bufA[q][e] = make_float2(pre[e], pim[e]);
        else          bufA[q][e] = make_float2(pre[256 - e], -pim[256 - e]);
    }
    __syncthreads();
    fft256(&bufA[q][0], &bufB[q][0], j, +1.0f);
    const int o = (int)(bc & 31);
    const float bo = bias[o];
    float* dst = out + row * Wn;
#pragma unroll
    for (int k = 0; k < 4; ++k) {
        int e = j + 64 * k;
        dst[e] = bufA[q][e].x * (1.0f / 256.0f) + bo;
    }
}

// ---------------------------------------------------------------------------
extern "C" void kernel_launch(void* const* d_in, const int* in_sizes, int n_in,
                              void* d_out, int out_size, void* d_ws, size_t ws_size,
                              hipStream_t stream) {
    (void)in_sizes; (void)n_in; (void)out_size; (void)ws_size;
    const float* x    = (const float*)d_in[0];
    const float* kre  = (const float*)d_in[1];
    const float* kim  = (const float*)d_in[2];
    const float* bias = (const float*)d_in[3];
    float* out = (float*)d_out;

    const long plane = (long)Bn * CIN * HWF;            // 8,454,144 floats
    float* Xre = (float*)d_ws;
    float* Xim = Xre + plane;

    // 1) row rfft: B*CIN*H / 4 rows-per-block
    rfft_rows<<<(Bn * CIN * Hn) / 4, 256, 0, stream>>>(x, Xre, Xim);
    // 2) column fft (forward, in place)
    fft_cols<<<Bn * CIN * 33, 256, 0, stream>>>(Xre, Xim, -1.0f, 1.0f);
    // 3) spectral channel mix (WMMA f32), Y overwrites X in place
    spectral_mix<<<HWF / 4, 256, 0, stream>>>(kre, kim, Xre, Xim);
    // 4) column ifft (inverse, in place, x 1/256)
    fft_cols<<<Bn * COUT * 33, 256, 0, stream>>>(Xre, Xim, +1.0f, 1.0f / 256.0f);
    // 5) row irfft + remaining 1/256 + bias
    irfft_rows_bias<<<(Bn * COUT * Hn) / 4, 256, 0, stream>>>(Xre, Xim, bias, out);
}